// RHMM_14104672600494
// MI455X (gfx1250) — compile-verified
//
#include <hip/hip_runtime.h>
#include <math.h>

#define Bdim 4
#define Tdim 256
#define Ndim 4096
#define Kdim 64
#define NNZt (Bdim * Ndim * 4)   // 65536 edges per step
#define GBLOCKS 64
#define NTHREADS 256
#define BLK_PER_B (GBLOCKS / Bdim) // 16

typedef __attribute__((ext_vector_type(2))) float v2f;
typedef __attribute__((ext_vector_type(8))) float v8f;

// ---- workspace byte offsets ----
// [0,64)       : control ints (cnt, gen) -- zeroed via hipMemsetAsync each launch
// [256,..)     : Mpart[64]   per-block la-max partials
// [768,..)     : Sout[4]     output exp-sum accumulators
// [1024,..)    : ssum[16384] per-state scatter accumulators
// [262144,..)  : em[B*T*N]   log-softmaxed emissions (16 MB)
#define OFF_MPART 256
#define OFF_SOUT  768
#define OFF_SSUM  1024
#define OFF_EM    262144

// ---------------- emissions GEMM: (1024x64) x (64x4096), fp32 WMMA ----------------
__global__ void __launch_bounds__(NTHREADS) emis_gemm(const float* __restrict__ A,
                                                      const float* __restrict__ W,
                                                      float* __restrict__ out) {
  const int lane = threadIdx.x & 31;
  const int wave = threadIdx.x >> 5;     // 8 waves, each a 16-wide N tile
  const int l15  = lane & 15;
  const int lh   = lane >> 4;            // half-wave select
  const int m0   = blockIdx.x * 16;
  const int n0   = blockIdx.y * 128 + wave * 16;
#if __has_builtin(__builtin_amdgcn_wmma_f32_16x16x4_f32)
  v8f c = {};
  // A frag: lane row = m0+l15, K pair = 4*kk + 2*lh + {0,1}
  const float* ap = A + (size_t)(m0 + l15) * Kdim + 2 * lh;
  // B frag: rows 4*kk + 2*lh + {0,1}, col n0+l15
  const float* bp = W + (size_t)(2 * lh) * Ndim + n0 + l15;
#pragma unroll
  for (int kk = 0; kk < Kdim / 4; ++kk) {
    v2f a, b;
    a.x = ap[4 * kk + 0];
    a.y = ap[4 * kk + 1];
    b.x = bp[(size_t)(4 * kk) * Ndim];
    b.y = bp[(size_t)(4 * kk + 1) * Ndim];
    c = __builtin_amdgcn_wmma_f32_16x16x4_f32(false, a, false, b, (short)0, c,
                                              false, false);
  }
  const int row0 = m0 + 8 * lh;   // VGPR v holds M = v (+8 for upper half-wave)
  const int col  = n0 + l15;
#pragma unroll
  for (int v = 0; v < 8; ++v)
    out[(size_t)(row0 + v) * Ndim + col] = c[v];
#else
  // scalar fallback (should not trigger on gfx1250)
  for (int idx = threadIdx.x; idx < 16 * 128; idx += NTHREADS) {
    int r = idx >> 7, cc = idx & 127;
    int row = m0 + r, cl = blockIdx.y * 128 + cc;
    float acc = 0.f;
    for (int k = 0; k < Kdim; ++k)
      acc += A[(size_t)row * Kdim + k] * W[(size_t)k * Ndim + cl];
    out[(size_t)row * Ndim + cl] = acc;
  }
#endif
}

// ---------------- in-place row log-softmax over N=4096 ----------------
__global__ void __launch_bounds__(NTHREADS) row_logsm(float* __restrict__ em) {
  float* p = em + (size_t)blockIdx.x * Ndim;
  __shared__ float red[NTHREADS];
  const int tid = threadIdx.x;
  float mx = -INFINITY;
  for (int i = tid; i < Ndim; i += NTHREADS) mx = fmaxf(mx, p[i]);
  red[tid] = mx; __syncthreads();
  for (int s = NTHREADS / 2; s > 0; s >>= 1) {
    if (tid < s) red[tid] = fmaxf(red[tid], red[tid + s]);
    __syncthreads();
  }
  mx = red[0]; __syncthreads();
  float sm = 0.f;
  for (int i = tid; i < Ndim; i += NTHREADS) sm += __expf(p[i] - mx);
  red[tid] = sm; __syncthreads();
  for (int s = NTHREADS / 2; s > 0; s >>= 1) {
    if (tid < s) red[tid] += red[tid + s];
    __syncthreads();
  }
  const float lse = mx + __logf(red[0]);
  for (int i = tid; i < Ndim; i += NTHREADS) p[i] -= lse;
}

// ---------------- grid-wide barrier (persistent kernel) ----------------
__device__ __forceinline__ void gsync(int* cnt, int* gen) {
  __syncthreads();
  if (threadIdx.x == 0) {
    __threadfence();
    int g = __hip_atomic_load(gen, __ATOMIC_ACQUIRE, __HIP_MEMORY_SCOPE_AGENT);
    if (atomicAdd(cnt, 1) == GBLOCKS - 1) {
      __hip_atomic_store(cnt, 0, __ATOMIC_RELAXED, __HIP_MEMORY_SCOPE_AGENT);
      __hip_atomic_fetch_add(gen, 1, __ATOMIC_RELEASE, __HIP_MEMORY_SCOPE_AGENT);
    } else {
      while (__hip_atomic_load(gen, __ATOMIC_ACQUIRE, __HIP_MEMORY_SCOPE_AGENT) == g)
        __builtin_amdgcn_s_sleep(1);
    }
  }
  __syncthreads();
}

// ---------------- persistent forward scan: 1 thread == 1 state ----------------
__global__ void __launch_bounds__(NTHREADS) scan_persist(
    const float* __restrict__ em, const int* __restrict__ duration,
    const int* __restrict__ tidx, const float* __restrict__ tlv,
    float* __restrict__ ssum, float* __restrict__ Mpart,
    float* __restrict__ Sout, int* cnt, int* gen, float* __restrict__ out) {
  const int tid = threadIdx.x;
  const int gid = blockIdx.x * NTHREADS + tid;  // state id in [0, B*N)
  const int b   = gid >> 12;                    // batch (4096 states per b)
  const int n   = gid & (Ndim - 1);
  const int dur = duration[b];

  __shared__ float red[NTHREADS];
  __shared__ float sM[Bdim];

  // ---- t = 0 init: la0 kept in a register ----
  float myla = em[(size_t)(b * Tdim) * Ndim + n] - __logf((float)Ndim);
  ssum[gid] = 0.f;
  if (gid < Bdim) Sout[gid] = 0.f;
  red[tid] = myla; __syncthreads();
  for (int s = NTHREADS / 2; s > 0; s >>= 1) {
    if (tid < s) red[tid] = fmaxf(red[tid], red[tid + s]);
    __syncthreads();
  }
  if (tid == 0) Mpart[blockIdx.x] = red[0];
  gsync(cnt, gen);

  for (int t = 1; t < Tdim; ++t) {
    // per-batch max of la(t-1) from 16 block partials
    if (tid < Bdim) {
      float m = -INFINITY;
#pragma unroll
      for (int k = 0; k < BLK_PER_B; ++k)
        m = fmaxf(m, Mpart[tid * BLK_PER_B + k]);
      sM[tid] = m;
    }
    __syncthreads();
    const float Mb = sM[b];

    // fold logsumexp-over-N output for time t-1 if it is this batch's duration
    if (dur == t) {
      red[tid] = __expf(myla - Mb); __syncthreads();
      for (int s = NTHREADS / 2; s > 0; s >>= 1) {
        if (tid < s) red[tid] += red[tid + s];
        __syncthreads();
      }
      if (tid == 0) unsafeAtomicAdd(&Sout[b], red[0]);
      __syncthreads();
    }

    // edge pass: this state's 4 outgoing edges (src==gid by construction)
    {
      const size_t eoff = (size_t)(t - 1) * NNZt + (size_t)gid * 4;
      const int4* qp = (const int4*)(tidx + eoff * 3);   // 48B, 16B-aligned
      const float4 lv = *(const float4*)(tlv + eoff);
      const int4 q0 = qp[0], q1 = qp[1], q2 = qp[2];     // tgt at .z,.y,.x,.w
      const float base = myla - Mb;
      float* sb = ssum + (b << 12);
      unsafeAtomicAdd(sb + q0.z, __expf(base + lv.x));
      unsafeAtomicAdd(sb + q1.y, __expf(base + lv.y));
      unsafeAtomicAdd(sb + q2.x, __expf(base + lv.z));
      unsafeAtomicAdd(sb + q2.w, __expf(base + lv.w));
      if (t < Tdim - 1) {  // prefetch next step's edge stream across the barrier
        __builtin_prefetch((const void*)(qp + 3 * (NNZt / 4)), 0, 1);
        __builtin_prefetch((const void*)(tlv + eoff + NNZt), 0, 1);
      }
    }
    gsync(cnt, gen);

    // finalize: la(t) = log(sum exp) + Mb + em[:,t,:]
    const float s = ssum[gid];
    const float lanew =
        (s > 0.f) ? (__logf(s) + Mb + em[((size_t)b * Tdim + t) * Ndim + n])
                  : -INFINITY;
    ssum[gid] = 0.f;
    myla = lanew;
    red[tid] = lanew; __syncthreads();
    for (int s2 = NTHREADS / 2; s2 > 0; s2 >>= 1) {
      if (tid < s2) red[tid] = fmaxf(red[tid], red[tid + s2]);
      __syncthreads();
    }
    if (tid == 0) Mpart[blockIdx.x] = red[0];
    gsync(cnt, gen);
  }

  // epilogue: time index T-1 (duration == T)
  if (tid < Bdim) {
    float m = -INFINITY;
#pragma unroll
    for (int k = 0; k < BLK_PER_B; ++k)
      m = fmaxf(m, Mpart[tid * BLK_PER_B + k]);
    sM[tid] = m;
  }
  __syncthreads();
  const float MbF = sM[b];
  if (dur == Tdim) {
    red[tid] = __expf(myla - MbF); __syncthreads();
    for (int s = NTHREADS / 2; s > 0; s >>= 1) {
      if (tid < s) red[tid] += red[tid + s];
      __syncthreads();
    }
    if (tid == 0) unsafeAtomicAdd(&Sout[b], red[0]);
  }
  gsync(cnt, gen);
  if (gid < Bdim) {
    float m = -INFINITY;
    for (int k = 0; k < BLK_PER_B; ++k)
      m = fmaxf(m, Mpart[gid * BLK_PER_B + k]);
    out[gid] = __logf(Sout[gid]) + m;   // log_probs[b, 0]
  }
}

extern "C" void kernel_launch(void* const* d_in, const int* in_sizes, int n_in,
                              void* d_out, int out_size, void* d_ws,
                              size_t ws_size, hipStream_t stream) {
  const float* obs      = (const float*)d_in[0];  // (B,T,K) fp32
  const float* wem      = (const float*)d_in[1];  // (K,N)   fp32
  const int*   duration = (const int*)d_in[2];    // (B,)
  const int*   tidx     = (const int*)d_in[3];    // (L,NNZ,3)
  const float* tlv      = (const float*)d_in[4];  // (L,NNZ)
  float* out = (float*)d_out;

  char* ws = (char*)d_ws;
  int*   cnt   = (int*)ws;
  int*   gen   = (int*)(ws + 4);
  float* Mpart = (float*)(ws + OFF_MPART);
  float* Sout  = (float*)(ws + OFF_SOUT);
  float* ssum  = (float*)(ws + OFF_SSUM);
  float* em    = (float*)(ws + OFF_EM);

  hipMemsetAsync(ws, 0, 64, stream);  // reset barrier state every launch

  emis_gemm<<<dim3((Bdim * Tdim) / 16, Ndim / 128), NTHREADS, 0, stream>>>(
      obs, wem, em);
  row_logsm<<<Bdim * Tdim, NTHREADS, 0, stream>>>(em);
  scan_persist<<<GBLOCKS, NTHREADS, 0, stream>>>(em, duration, tidx, tlv, ssum,
                                                 Mpart, Sout, cnt, gen, out);
}